// InformerEncoderLayer_20186346291966
// MI455X (gfx1250) — compile-verified
//
#include <hip/hip_runtime.h>
#include <hip/hip_bf16.h>
#include <math.h>

typedef __attribute__((ext_vector_type(2))) float v2f;
typedef __attribute__((ext_vector_type(8))) float v8f;
typedef __attribute__((ext_vector_type(4))) unsigned v4u;
typedef __attribute__((ext_vector_type(8))) unsigned v8u;

#define BATCH 2
#define SEQ   2048
#define CDIM  512
#define HEADS 8
#define DKH   64
#define FFDIM 2048
#define NPOOL 1024
#define NTOP  512
#define KC    8          // K-chunk staged per TDM transfer

static __device__ __forceinline__ v8f wmma4(v2f a, v2f b, v8f c) {
    // V_WMMA_F32_16X16X4_F32 : D = A(16x4,f32) * B(4x16,f32) + C(16x16,f32)
    return __builtin_amdgcn_wmma_f32_16x16x4_f32(false, a, false, b, (short)0, c, false, false);
}

// Issue one Tensor-Data-Mover descriptor: DMA a [rows x width] f32 tile
// (row stride = stride_elems) from global into LDS at lds_addr (bytes).
// Uniform (SGPR) operands; tracked on TENSORcnt of the issuing wave.
static __device__ __forceinline__ void tdm_load_2d(unsigned lds_addr, const float* gptr,
                                                   int width, int rows, int stride_elems)
{
    unsigned long long ga = (unsigned long long)(uintptr_t)gptr;
    v4u g0;
    g0[0] = 1u;                                            // count=1, user mode, no gather
    g0[1] = lds_addr;                                      // LDS byte address
    g0[2] = (unsigned)(ga & 0xffffffffu);                  // global_addr[31:0]
    g0[3] = ((unsigned)(ga >> 32) & 0x01ffffffu) | 0x80000000u; // addr[56:32] | type=2
    v8u g1;
    g1[0] = 2u << 16;                                      // data_size = 4 bytes
    g1[1] = ((unsigned)width & 0xffffu) << 16;             // tensor_dim0[15:0]
    g1[2] = (((unsigned)width >> 16) & 0xffffu)            // tensor_dim0[31:16]
          | (((unsigned)rows & 0xffffu) << 16);            // tensor_dim1[15:0]
    g1[3] = (((unsigned)rows >> 16) & 0xffffu)             // tensor_dim1[31:16]
          | (((unsigned)width & 0xffffu) << 16);           // tile_dim0
    g1[4] = (unsigned)rows & 0xffffu;                      // tile_dim1 (tile_dim2=0 -> 2D)
    g1[5] = (unsigned)stride_elems;                        // tensor_dim0_stride[31:0]
    g1[6] = 0u;                                            // stride hi / dim1_stride
    g1[7] = 0u;
    asm volatile("tensor_load_to_lds %0, %1" :: "s"(g0), "s"(g1) : "memory");
}

// ---------------------------------------------------------------------------
// WMMA GEMM with TDM-fed, double-buffered LDS staging.
// out[M,Nc] = A[M,K] @ W[K,Nc] + bias (optional exact GELU).
// block = 256 (8 waves); block covers rows [by*16,+16) x cols [bx*512,+512);
// wave w owns the 64-col stripe w*64.
// ---------------------------------------------------------------------------
template<int K, int Nc, int ACT>
__global__ void gemm_wmma_kernel(const float* __restrict__ A, const float* __restrict__ W,
                                 const float* __restrict__ bias, float* __restrict__ out)
{
    __shared__ float Wlds[2][KC * 512];    // 2 x 16 KB
    __shared__ float Alds[2][16 * KC];     // 2 x 512 B
    const int lane = threadIdx.x & 31;
    const int wave = threadIdx.x >> 5;
    const int n15  = lane & 15;
    const int hi   = lane >> 4;
    const int ksub = hi << 1;
    const int row0 = blockIdx.y * 16;
    const int col0 = blockIdx.x * 512;
    const int wcol = wave * 64 + n15;

    v8f acc[4];
    #pragma unroll
    for (int t = 0; t < 4; ++t)
        #pragma unroll
        for (int r = 0; r < 8; ++r) acc[t][r] = 0.0f;

    const int NCHUNK = K / KC;
    if (wave == 0) {
        tdm_load_2d((unsigned)(uintptr_t)&Wlds[0][0], W + (size_t)0 * Nc + col0, 512, KC, Nc);
        tdm_load_2d((unsigned)(uintptr_t)&Alds[0][0], A + (size_t)row0 * K + 0, KC, 16, K);
    }

    #pragma unroll 1
    for (int c = 0; c < NCHUNK; ++c) {
        if (wave == 0) __builtin_amdgcn_s_wait_tensorcnt(0);   // chunk c resident in LDS
        __syncthreads();                                       // publish; also WAR fence for buf reuse
        if (wave == 0 && c + 1 < NCHUNK) {
            const int k1 = (c + 1) * KC;
            tdm_load_2d((unsigned)(uintptr_t)&Wlds[(c + 1) & 1][0],
                        W + (size_t)k1 * Nc + col0, 512, KC, Nc);
            tdm_load_2d((unsigned)(uintptr_t)&Alds[(c + 1) & 1][0],
                        A + (size_t)row0 * K + k1, KC, 16, K);
        }
        const float* Wb = &Wlds[c & 1][0];
        const float* Ab = &Alds[c & 1][0];
        #pragma unroll
        for (int kk = 0; kk < KC; kk += 4) {
            v2f a = *(const v2f*)(Ab + n15 * KC + kk + ksub);
            #pragma unroll
            for (int t = 0; t < 4; ++t) {
                v2f b;
                b.x = Wb[(kk + ksub) * 512 + wcol + t * 16];
                b.y = Wb[(kk + ksub + 1) * 512 + wcol + t * 16];
                acc[t] = wmma4(a, b, acc[t]);
            }
        }
    }

    #pragma unroll
    for (int t = 0; t < 4; ++t) {
        const int col = col0 + wave * 64 + t * 16 + n15;
        const float bv = bias[col];
        #pragma unroll
        for (int r = 0; r < 8; ++r) {
            const int row = row0 + r + (hi << 3);
            float v = acc[t][r] + bv;
            if (ACT == 1) v = 0.5f * v * (1.0f + erff(v * 0.70710678118654752f));
            out[(size_t)row * Nc + col] = v;
        }
    }
}

// ---------------------------------------------------------------------------
// Attention pass A: per (b,h,16-row q tile) stream score tiles through WMMA,
// keep per-lane running row max / row sum, butterfly-reduce at the end.
// grid = (SEQ/16, HEADS, BATCH), block = 32 (one wave).
// ---------------------------------------------------------------------------
__global__ void attn_stats_kernel(const float* __restrict__ qkv,
                                  float* __restrict__ rowmax, float* __restrict__ rowsum)
{
    const int lane = threadIdx.x & 31;
    const int n15  = lane & 15;
    const int hi   = lane >> 4;
    const int ksub = hi << 1;
    const int nt = blockIdx.x, h = blockIdx.y, b = blockIdx.z;

    const float* qbase = qkv + (size_t)(b * SEQ + nt * 16 + n15) * (3 * CDIM) + h * DKH;
    float qreg[32];
    #pragma unroll
    for (int kb = 0; kb < 16; ++kb) {
        v2f qq = *(const v2f*)(qbase + 4 * kb + ksub);
        qreg[2 * kb] = qq.x; qreg[2 * kb + 1] = qq.y;
    }

    float pmax[8], psum[8];
    #pragma unroll
    for (int r = 0; r < 8; ++r) { pmax[r] = -3.0e38f; psum[r] = 0.0f; }
    const float scale = 0.125f;  // 1/sqrt(64)

    for (int mt = 0; mt < SEQ / 16; ++mt) {
        const float* kbase = qkv + (size_t)(b * SEQ + mt * 16 + n15) * (3 * CDIM) + CDIM + h * DKH;
        v8f c;
        #pragma unroll
        for (int r = 0; r < 8; ++r) c[r] = 0.0f;
        #pragma unroll
        for (int kb = 0; kb < 16; ++kb) {
            v2f a; a.x = qreg[2 * kb]; a.y = qreg[2 * kb + 1];
            v2f bb = *(const v2f*)(kbase + 4 * kb + ksub);
            c = wmma4(a, bb, c);
        }
        #pragma unroll
        for (int r = 0; r < 8; ++r) {
            float s = c[r] * scale;
            pmax[r] = fmaxf(pmax[r], s);
            psum[r] += s;
        }
    }
    #pragma unroll
    for (int m = 1; m < 16; m <<= 1) {
        #pragma unroll
        for (int r = 0; r < 8; ++r) {
            pmax[r] = fmaxf(pmax[r], __shfl_xor(pmax[r], m, 32));
            psum[r] += __shfl_xor(psum[r], m, 32);
        }
    }
    if (n15 == 0) {
        #pragma unroll
        for (int r = 0; r < 8; ++r) {
            const int row = nt * 16 + r + (hi << 3);
            rowmax[(size_t)(b * HEADS + h) * SEQ + row] = pmax[r];
            rowsum[(size_t)(b * HEADS + h) * SEQ + row] = psum[r];
        }
    }
}

// Rank-count top-k (JAX tie-break: smaller index wins).
__global__ void topk_sel_kernel(const float* __restrict__ rowmax, const float* __restrict__ rowsum,
                                int* __restrict__ sel)
{
    const int idx = blockIdx.x * blockDim.x + threadIdx.x;
    if (idx >= BATCH * HEADS * SEQ) return;
    const int bh = idx / SEQ, n = idx % SEQ;
    const float* rm = rowmax + (size_t)bh * SEQ;
    const float* rs = rowsum + (size_t)bh * SEQ;
    const float inv = 1.0f / (float)SEQ;
    const float Mn = rm[n] - rs[n] * inv;
    int rank = 0;
    for (int m = 0; m < SEQ; ++m) {
        const float Mm = rm[m] - rs[m] * inv;
        rank += (Mm > Mn) || (Mm == Mn && m < n);
    }
    sel[idx] = (rank < NTOP) ? 1 : 0;
}

// mean of V over sequence per (b,h,d) — output for masked-out rows.
__global__ void vmean_kernel(const float* __restrict__ qkv, float* __restrict__ vmean)
{
    const int d = threadIdx.x;               // 64
    const int h = blockIdx.x, b = blockIdx.y;
    float s = 0.0f;
    for (int n = 0; n < SEQ; ++n)
        s += qkv[(size_t)(b * SEQ + n) * (3 * CDIM) + 2 * CDIM + h * DKH + d];
    vmean[(b * HEADS + h) * DKH + d] = s * (1.0f / (float)SEQ);
}

// ---------------------------------------------------------------------------
// Attention pass B: recompute score tiles (WMMA), softmax with known rowmax,
// P-tile via LDS back into WMMA A-layout, accumulate ctx = P @ V (WMMA).
// grid = (SEQ/16, HEADS, BATCH), block = 32 (one wave).
// ---------------------------------------------------------------------------
__global__ void attn_ctx_kernel(const float* __restrict__ qkv, const float* __restrict__ rowmax,
                                const int* __restrict__ sel, const float* __restrict__ vmean,
                                float* __restrict__ ctx)
{
    __shared__ float P[16 * 17];
    const int lane = threadIdx.x & 31;
    const int n15  = lane & 15;
    const int hi   = lane >> 4;
    const int ksub = hi << 1;
    const int nt = blockIdx.x, h = blockIdx.y, b = blockIdx.z;

    const float* qbase = qkv + (size_t)(b * SEQ + nt * 16 + n15) * (3 * CDIM) + h * DKH;
    float qreg[32];
    #pragma unroll
    for (int kb = 0; kb < 16; ++kb) {
        v2f qq = *(const v2f*)(qbase + 4 * kb + ksub);
        qreg[2 * kb] = qq.x; qreg[2 * kb + 1] = qq.y;
    }
    float rmax[8];
    #pragma unroll
    for (int r = 0; r < 8; ++r)
        rmax[r] = rowmax[(size_t)(b * HEADS + h) * SEQ + nt * 16 + r + (hi << 3)];

    v8f acc[4];
    #pragma unroll
    for (int t = 0; t < 4; ++t)
        #pragma unroll
        for (int r = 0; r < 8; ++r) acc[t][r] = 0.0f;
    float pden[8];
    #pragma unroll
    for (int r = 0; r < 8; ++r) pden[r] = 0.0f;
    const float scale = 0.125f;

    for (int mt = 0; mt < SEQ / 16; ++mt) {
        // scores tile S = Q Kt
        const float* kbase = qkv + (size_t)(b * SEQ + mt * 16 + n15) * (3 * CDIM) + CDIM + h * DKH;
        v8f c;
        #pragma unroll
        for (int r = 0; r < 8; ++r) c[r] = 0.0f;
        #pragma unroll
        for (int kb = 0; kb < 16; ++kb) {
            v2f a; a.x = qreg[2 * kb]; a.y = qreg[2 * kb + 1];
            v2f bb = *(const v2f*)(kbase + 4 * kb + ksub);
            c = wmma4(a, bb, c);
        }
        // P = exp(S - rowmax)  (accumulator layout) -> LDS
        #pragma unroll
        for (int r = 0; r < 8; ++r) {
            float p = __expf(c[r] * scale - rmax[r]);
            pden[r] += p;
            P[(r + (hi << 3)) * 17 + n15] = p;
        }
        asm volatile("s_wait_dscnt 0x0" ::: "memory");  // same-wave LDS RAW fence
        // ctx += P(16x16) @ V(16x64) : 4 K-steps x 4 d-tiles
        #pragma unroll
        for (int kk = 0; kk < 16; kk += 4) {
            v2f a;
            a.x = P[n15 * 17 + kk + ksub];
            a.y = P[n15 * 17 + kk + ksub + 1];
            const float* vb0 = qkv + (size_t)(b * SEQ + mt * 16 + kk + ksub) * (3 * CDIM)
                             + 2 * CDIM + h * DKH + n15;
            const float* vb1 = vb0 + 3 * CDIM;
            #pragma unroll
            for (int t = 0; t < 4; ++t) {
                v2f bb; bb.x = vb0[t * 16]; bb.y = vb1[t * 16];
                acc[t] = wmma4(a, bb, acc[t]);
            }
        }
    }
    // softmax denominator: reduce across the 16 lanes of each half (all lanes get it)
    #pragma unroll
    for (int m = 1; m < 16; m <<= 1)
        #pragma unroll
        for (int r = 0; r < 8; ++r) pden[r] += __shfl_xor(pden[r], m, 32);

    #pragma unroll
    for (int r = 0; r < 8; ++r) {
        const int row = nt * 16 + r + (hi << 3);
        const int sl  = sel[(size_t)(b * HEADS + h) * SEQ + row];
        const float dinv = 1.0f / pden[r];
        #pragma unroll
        for (int t = 0; t < 4; ++t) {
            const int d = t * 16 + n15;
            const float val = sl ? acc[t][r] * dinv : vmean[(b * HEADS + h) * DKH + d];
            ctx[(size_t)(b * SEQ + row) * CDIM + h * DKH + d] = val;
        }
    }
}

// Residual + LayerNorm over C=512. block = 128, one row per block.
__global__ void ln_res_kernel(const float* __restrict__ x, const float* __restrict__ add,
                              const float* __restrict__ g, const float* __restrict__ be,
                              float* __restrict__ out)
{
    const int row = blockIdx.x, tid = threadIdx.x;
    const float* xr = x + (size_t)row * CDIM;
    const float* ar = add + (size_t)row * CDIM;
    float vals[4], lsum = 0.0f, lsq = 0.0f;
    #pragma unroll
    for (int i = 0; i < 4; ++i) {
        const float v = xr[tid + i * 128] + ar[tid + i * 128];
        vals[i] = v; lsum += v; lsq += v * v;
    }
    #pragma unroll
    for (int m = 1; m < 32; m <<= 1) { lsum += __shfl_xor(lsum, m, 32); lsq += __shfl_xor(lsq, m, 32); }
    __shared__ float s1[4], s2[4];
    if ((tid & 31) == 0) { s1[tid >> 5] = lsum; s2[tid >> 5] = lsq; }
    __syncthreads();
    const float ts = s1[0] + s1[1] + s1[2] + s1[3];
    const float tq = s2[0] + s2[1] + s2[2] + s2[3];
    const float mu = ts * (1.0f / CDIM);
    const float inv = rsqrtf(tq * (1.0f / CDIM) - mu * mu + 1e-5f);
    #pragma unroll
    for (int i = 0; i < 4; ++i) {
        const int c = tid + i * 128;
        out[(size_t)row * CDIM + c] = (vals[i] - mu) * inv * g[c] + be[c];
    }
}

// Pack conv weights [O=512, I=512, T=3] -> [T*512 + I, O]
__global__ void wpack_kernel(const float* __restrict__ cw, float* __restrict__ wconv)
{
    const int idx = blockIdx.x * blockDim.x + threadIdx.x;
    if (idx >= 3 * CDIM * CDIM) return;
    const int o = idx % CDIM;
    const int r = idx / CDIM;
    const int i = r % CDIM;
    const int t = r / CDIM;
    wconv[idx] = cw[((size_t)o * CDIM + i) * 3 + t];
}

// im2col with circular pad-1: col[(b*N+n)*1536 + t*512 + i] = x[b, (n-1+t) mod N, i]
__global__ void im2col_kernel(const float* __restrict__ x, float* __restrict__ col)
{
    const size_t idx = (size_t)blockIdx.x * blockDim.x + threadIdx.x;
    if (idx >= (size_t)BATCH * SEQ * 3 * CDIM) return;
    const int i = (int)(idx % CDIM);
    const int t = (int)((idx / CDIM) % 3);
    const size_t bn = idx / (3 * CDIM);
    const int n = (int)(bn % SEQ);
    const int b = (int)(bn / SEQ);
    const int nn = (n - 1 + t + SEQ) & (SEQ - 1);
    col[idx] = x[((size_t)(b * SEQ + nn)) * CDIM + i];
}

// maxpool(k3,s2,p1) -> ELU -> LayerNorm, fused. block = 128, one out-row per block.
__global__ void pool_elu_ln_kernel(const float* __restrict__ conv, const float* __restrict__ g,
                                   const float* __restrict__ be, float* __restrict__ out)
{
    const int row = blockIdx.x;               // b*NPOOL + j
    const int b = row >> 10, j = row & (NPOOL - 1);
    const int tid = threadIdx.x;
    float vals[4], lsum = 0.0f, lsq = 0.0f;
    #pragma unroll
    for (int i = 0; i < 4; ++i) {
        const int c = tid + i * 128;
        float m = -3.0e38f;
        #pragma unroll
        for (int t = -1; t <= 1; ++t) {
            const int n = 2 * j + t;
            if (n >= 0 && n < SEQ)
                m = fmaxf(m, conv[((size_t)(b * SEQ + n)) * CDIM + c]);
        }
        const float v = (m > 0.0f) ? m : (__expf(m) - 1.0f);   // ELU(alpha=1)
        vals[i] = v; lsum += v; lsq += v * v;
    }
    #pragma unroll
    for (int m = 1; m < 32; m <<= 1) { lsum += __shfl_xor(lsum, m, 32); lsq += __shfl_xor(lsq, m, 32); }
    __shared__ float s1[4], s2[4];
    if ((tid & 31) == 0) { s1[tid >> 5] = lsum; s2[tid >> 5] = lsq; }
    __syncthreads();
    const float ts = s1[0] + s1[1] + s1[2] + s1[3];
    const float tq = s2[0] + s2[1] + s2[2] + s2[3];
    const float mu = ts * (1.0f / CDIM);
    const float inv = rsqrtf(tq * (1.0f / CDIM) - mu * mu + 1e-5f);
    #pragma unroll
    for (int i = 0; i < 4; ++i) {
        const int c = tid + i * 128;
        out[(size_t)row * CDIM + c] = (vals[i] - mu) * inv * g[c] + be[c];
    }
}

// ---------------------------------------------------------------------------
extern "C" void kernel_launch(void* const* d_in, const int* in_sizes, int n_in,
                              void* d_out, int out_size, void* d_ws, size_t ws_size,
                              hipStream_t stream)
{
    const float* x      = (const float*)d_in[0];
    const float* qkv_w  = (const float*)d_in[1];
    const float* qkv_b  = (const float*)d_in[2];
    const float* out_w  = (const float*)d_in[3];
    const float* out_b  = (const float*)d_in[4];
    const float* ffn_w1 = (const float*)d_in[5];
    const float* ffn_b1 = (const float*)d_in[6];
    const float* ffn_w2 = (const float*)d_in[7];
    const float* ffn_b2 = (const float*)d_in[8];
    const float* n1_g   = (const float*)d_in[9];
    const float* n1_b   = (const float*)d_in[10];
    const float* n2_g   = (const float*)d_in[11];
    const float* n2_b   = (const float*)d_in[12];
    const float* conv_w = (const float*)d_in[13];
    const float* conv_b = (const float*)d_in[14];
    const float* cn_g   = (const float*)d_in[15];
    const float* cn_b   = (const float*)d_in[16];

    float* ws = (float*)d_ws;
    const int ROWS = BATCH * SEQ;                    // 4096
    size_t o = 0;
    float* qkv    = ws;                 o += (size_t)ROWS * 3 * CDIM;   // 6,291,456
    float* col    = qkv;                                                // alias (qkv dead by then)
    float* rowmax = ws + o;             o += BATCH * HEADS * SEQ;
    float* rowsum = ws + o;             o += BATCH * HEADS * SEQ;
    int*   sel    = (int*)(ws + o);     o += BATCH * HEADS * SEQ;
    float* vmean  = ws + o;             o += BATCH * HEADS * DKH;
    float* ctx    = ws + o;             o += (size_t)ROWS * CDIM;
    float* convo  = ctx;                                                // alias (ctx dead by then)
    float* attnout= ws + o;             o += (size_t)ROWS * CDIM;       // also reused for ffn_out
    float* x1     = ws + o;             o += (size_t)ROWS * CDIM;
    float* hbuf   = ws + o;             o += (size_t)ROWS * FFDIM;
    float* x2     = ws + o;             o += (size_t)ROWS * CDIM;
    float* wconv  = ws + o;             o += (size_t)3 * CDIM * CDIM;
    (void)ws_size; (void)n_in; (void)in_sizes; (void)out_size;

    const dim3 blk256(256);

    // 1. QKV projection
    gemm_wmma_kernel<CDIM, 3 * CDIM, 0><<<dim3(3, ROWS / 16), blk256, 0, stream>>>(x, qkv_w, qkv_b, qkv);
    // 2. row max / mean of scaled scores
    attn_stats_kernel<<<dim3(SEQ / 16, HEADS, BATCH), 32, 0, stream>>>(qkv, rowmax, rowsum);
    // 3. top-k row selection + V mean
    topk_sel_kernel<<<dim3((BATCH * HEADS * SEQ) / 256), blk256, 0, stream>>>(rowmax, rowsum, sel);
    vmean_kernel<<<dim3(HEADS, BATCH), 64, 0, stream>>>(qkv, vmean);
    // 4. context (softmax @ V with row selection)
    attn_ctx_kernel<<<dim3(SEQ / 16, HEADS, BATCH), 32, 0, stream>>>(qkv, rowmax, sel, vmean, ctx);
    // 5. output projection
    gemm_wmma_kernel<CDIM, CDIM, 0><<<dim3(1, ROWS / 16), blk256, 0, stream>>>(ctx, out_w, out_b, attnout);
    // 6. LN(x + attn)
    ln_res_kernel<<<dim3(ROWS), dim3(128), 0, stream>>>(x, attnout, n1_g, n1_b, x1);
    // 7. FFN
    gemm_wmma_kernel<CDIM, FFDIM, 1><<<dim3(4, ROWS / 16), blk256, 0, stream>>>(x1, ffn_w1, ffn_b1, hbuf);
    gemm_wmma_kernel<FFDIM, CDIM, 0><<<dim3(1, ROWS / 16), blk256, 0, stream>>>(hbuf, ffn_w2, ffn_b2, attnout);
    ln_res_kernel<<<dim3(ROWS), dim3(128), 0, stream>>>(x1, attnout, n2_g, n2_b, x2);
    // 8. distilling conv (circular, k=3) as im2col GEMM
    wpack_kernel<<<dim3((3 * CDIM * CDIM) / 256), blk256, 0, stream>>>(conv_w, wconv);
    im2col_kernel<<<dim3((ROWS * 3 * CDIM) / 256), blk256, 0, stream>>>(x2, col);
    gemm_wmma_kernel<3 * CDIM, CDIM, 0><<<dim3(1, ROWS / 16), blk256, 0, stream>>>(col, wconv, conv_b, convo);
    // 9. maxpool + ELU + LN -> output [B, 1024, 512]
    pool_elu_ln_kernel<<<dim3(BATCH * NPOOL), dim3(128), 0, stream>>>(convo, cn_g, cn_b, (float*)d_out);
}